// BERT_CON_27788438405734
// MI455X (gfx1250) — compile-verified
//
#include <hip/hip_runtime.h>

typedef __attribute__((ext_vector_type(2))) float v2f;
typedef __attribute__((ext_vector_type(8))) float v8f;
typedef __attribute__((ext_vector_type(4))) int   v4i;

#define BN     8192
#define DK     768
#define NC     12
#define LAMF   0.1f
#define GAMMAF 1.0f

#define TILE   128
#define KC     32
#define LDST   36              // padded LDS stride (floats): conflict-free + 16B aligned
#define NTILES (BN / TILE)     // 64
#define NSLABS (DK / KC)       // 24

#define AS1 __attribute__((address_space(1)))
#define AS3 __attribute__((address_space(3)))

#if defined(__AMDGCN__) && __has_builtin(__builtin_amdgcn_global_load_async_to_lds_b128)
#define USE_ASYNC 1
#else
#define USE_ASYNC 0
#endif

// ---------------------------------------------------------------------------
// Kernel 0: zero workspace accumulators
// ---------------------------------------------------------------------------
__global__ void init_kernel(float* ws) {
    if (threadIdx.x < 2) ws[threadIdx.x] = 0.0f;
}

// ---------------------------------------------------------------------------
// Kernel 1: sq[i] = ||E_i||^2   (one 32-lane group per row)
// ---------------------------------------------------------------------------
__global__ __launch_bounds__(256) void sq_kernel(const float* __restrict__ E,
                                                 float* __restrict__ sq) {
    __shared__ float part[256];
    int grp = threadIdx.x >> 5, lane = threadIdx.x & 31;
    int row = blockIdx.x * 8 + grp;
    const float* er = E + (size_t)row * DK;
    float s = 0.0f;
    for (int k = lane; k < DK; k += 32) { float v = er[k]; s += v * v; }
    part[threadIdx.x] = s;
    __syncthreads();
    if (lane == 0) {
        float tot = 0.0f;
        #pragma unroll
        for (int l = 0; l < 32; ++l) tot += part[grp * 32 + l];
        sq[row] = tot;
    }
}

// ---------------------------------------------------------------------------
// Kernel 2: fused Gram (E E^T via f32 WMMA) + contrastive-loss epilogue.
// Upper-triangular tile pairs only; off-diagonal tiles scaled by 2.
// 8 waves; each wave owns a 32x64 sub-tile (2 M-frags x 4 N-frags).
// K is slabbed through double-buffered LDS, filled by async global->LDS DMA.
// ---------------------------------------------------------------------------
__global__ __launch_bounds__(256) void gram_loss_kernel(const float* __restrict__ E,
                                                        const float* __restrict__ sq,
                                                        const int* __restrict__ labels,
                                                        float* __restrict__ closs_accum) {
    __shared__ float As[2][TILE * LDST];
    __shared__ float Bs[2][TILE * LDST];
    __shared__ float red[256];

    // decode linear block id -> upper-triangular (ti, tj), ti <= tj
    int rem = blockIdx.x;
    int ti = 0;
    while (rem >= (NTILES - ti)) { rem -= (NTILES - ti); ++ti; }
    int tj = ti + rem;

    const int i0 = ti * TILE;
    const int j0 = tj * TILE;

    const int tid  = threadIdx.x;
    const int wave = tid >> 5;
    const int lane = tid & 31;
    const int r    = lane & 15;      // row/col-within-16 for fragments
    const int h    = lane >> 4;      // K-half select (per ISA 16x4 f32 A layout)
    const int wm   = wave & 3;       // M strip: rows wm*32 .. +32
    const int wn   = wave >> 2;      // N strip: cols wn*64 .. +64

    v8f acc[2][4];
    const v8f vzero = {0.f, 0.f, 0.f, 0.f, 0.f, 0.f, 0.f, 0.f};
    #pragma unroll
    for (int mi = 0; mi < 2; ++mi)
        #pragma unroll
        for (int ni = 0; ni < 4; ++ni) acc[mi][ni] = vzero;

    // stage one 128xKC slab of A and B into LDS buffer `buf`
    auto stage = [&](int buf, int kb) {
        #pragma unroll
        for (int s2 = 0; s2 < 4; ++s2) {
            int q   = tid + 256 * s2;       // 0..1023 float4 slots
            int row = q >> 3;               // 8 float4 per 32-float row
            int c4  = (q & 7) << 2;
            const float* ga = E + (size_t)(i0 + row) * DK + kb + c4;
            const float* gb = E + (size_t)(j0 + row) * DK + kb + c4;
            float* la = &As[buf][row * LDST + c4];
            float* lb = &Bs[buf][row * LDST + c4];
#if USE_ASYNC
            __builtin_amdgcn_global_load_async_to_lds_b128(
                (AS1 v4i*)(uintptr_t)ga, (AS3 v4i*)la, 0, 0);
            __builtin_amdgcn_global_load_async_to_lds_b128(
                (AS1 v4i*)(uintptr_t)gb, (AS3 v4i*)lb, 0, 0);
#else
            *(float4*)la = *(const float4*)ga;
            *(float4*)lb = *(const float4*)gb;
#endif
        }
    };
    auto wait_fill = [&]() {
#if USE_ASYNC
#if __has_builtin(__builtin_amdgcn_s_wait_asynccnt)
        __builtin_amdgcn_s_wait_asynccnt(0);
#else
        asm volatile("s_wait_asynccnt 0x0" ::: "memory");
#endif
#endif
    };

    stage(0, 0);   // prologue fill

    const int arow0 = (wm * 32 + r) * LDST;
    const int arow1 = (wm * 32 + 16 + r) * LDST;
    const int brow  = (wn * 64 + r) * LDST;

    #pragma unroll 1
    for (int s = 0; s < NSLABS; ++s) {
        const int cur = s & 1;
        wait_fill();                 // my async DMA into buf[cur] is complete
        __syncthreads();             // everyone's DMA visible; prior compute done
        if (s + 1 < NSLABS) stage(cur ^ 1, (s + 1) * KC);

        const float* Ac = As[cur];
        const float* Bc = Bs[cur];
        #pragma unroll
        for (int k = 0; k < KC; k += 4) {
            v2f a0 = *(const v2f*)(Ac + arow0 + k + 2 * h);
            v2f a1 = *(const v2f*)(Ac + arow1 + k + 2 * h);
            #pragma unroll
            for (int ni = 0; ni < 4; ++ni) {
                v2f b = *(const v2f*)(Bc + brow + ni * 16 * LDST + k + 2 * h);
                acc[0][ni] = __builtin_amdgcn_wmma_f32_16x16x4_f32(
                    false, a0, false, b, (short)0, acc[0][ni], false, false);
                acc[1][ni] = __builtin_amdgcn_wmma_f32_16x16x4_f32(
                    false, a1, false, b, (short)0, acc[1][ni], false, false);
            }
        }
    }

    // epilogue: D layout -> lane holds (i = i0 + wm*32 + mi*16 + 8*h + g,
    //                                   j = j0 + wn*64 + ni*16 + r)
    const int ibase = i0 + wm * 32 + 8 * h;
    float sqi[2][8]; int li[2][8];
    #pragma unroll
    for (int mi = 0; mi < 2; ++mi)
        #pragma unroll
        for (int g = 0; g < 8; ++g) {
            int i = ibase + mi * 16 + g;
            sqi[mi][g] = sq[i];
            li[mi][g]  = labels[i];
        }

    float lsum = 0.0f;
    #pragma unroll
    for (int ni = 0; ni < 4; ++ni) {
        int j = j0 + wn * 64 + ni * 16 + r;
        float sqj = sq[j];
        int   lj  = labels[j];
        #pragma unroll
        for (int mi = 0; mi < 2; ++mi) {
            #pragma unroll
            for (int g = 0; g < 8; ++g) {
                int i = ibase + mi * 16 + g;
                float d2   = sqi[mi][g] + sqj - 2.0f * acc[mi][ni][g];
                float dist = sqrtf(fmaxf(d2, 0.0f));
                float per  = (li[mi][g] == lj) ? dist : fmaxf(0.0f, GAMMAF - dist);
                if (i != j) lsum += per;
            }
        }
    }
    if (ti != tj) lsum *= 2.0f;      // symmetric off-diagonal tiles counted once

    // block reduction -> one global atomic per workgroup
    red[tid] = lsum;
    __syncthreads();
    for (int s = 128; s > 0; s >>= 1) {
        if (tid < s) red[tid] += red[tid + s];
        __syncthreads();
    }
    if (tid == 0) atomicAdd(closs_accum, red[0]);
}

// ---------------------------------------------------------------------------
// Kernel 3: logits = X W^T + b, log-softmax, cross-entropy accumulation.
// One 128-thread block per row.
// ---------------------------------------------------------------------------
__global__ __launch_bounds__(128) void xent_kernel(const float* __restrict__ X,
                                                   const float* __restrict__ W,
                                                   const float* __restrict__ bias,
                                                   const int* __restrict__ labels,
                                                   float* __restrict__ logits_out,
                                                   float* __restrict__ cross_accum) {
    __shared__ float part[NC * 128];
    __shared__ float lg[NC];
    int row = blockIdx.x, t = threadIdx.x;
    const float* x = X + (size_t)row * DK;

    float a[NC];
    #pragma unroll
    for (int c = 0; c < NC; ++c) a[c] = 0.0f;
    for (int k = t; k < DK; k += 128) {
        float xv = x[k];
        #pragma unroll
        for (int c = 0; c < NC; ++c) a[c] += xv * W[c * DK + k];
    }
    #pragma unroll
    for (int c = 0; c < NC; ++c) part[c * 128 + t] = a[c];
    __syncthreads();

    if (t < NC) {
        float s = 0.0f;
        for (int l = 0; l < 128; ++l) s += part[t * 128 + l];
        s += bias[t];
        lg[t] = s;
        logits_out[(size_t)row * NC + t] = s;
    }
    __syncthreads();

    if (t == 0) {
        float m = lg[0];
        #pragma unroll
        for (int c = 1; c < NC; ++c) m = fmaxf(m, lg[c]);
        float se = 0.0f;
        #pragma unroll
        for (int c = 0; c < NC; ++c) se += expf(lg[c] - m);
        float lse = m + logf(se);
        int lbl = labels[row];
        float lp = lg[lbl] - lse;
        atomicAdd(cross_accum, -lp * (1.0f / (float)BN));
    }
}

// ---------------------------------------------------------------------------
// Kernel 4: combine -> loss = (1-lam)*CE + lam*contrastive_sum
// ---------------------------------------------------------------------------
__global__ void finalize_kernel(const float* __restrict__ ws, float* __restrict__ out) {
    if (threadIdx.x == 0) out[0] = (1.0f - LAMF) * ws[1] + LAMF * ws[0];
}

// ---------------------------------------------------------------------------
// Host entry
// ---------------------------------------------------------------------------
extern "C" void kernel_launch(void* const* d_in, const int* in_sizes, int n_in,
                              void* d_out, int out_size, void* d_ws, size_t ws_size,
                              hipStream_t stream) {
    const float* pooled = (const float*)d_in[0];   // [B, D]
    const float* target = (const float*)d_in[1];   // [B, D]
    const int*   labels = (const int*)d_in[2];     // [B]
    const float* W      = (const float*)d_in[3];   // [C, D]
    const float* bias   = (const float*)d_in[4];   // [C]

    float* out = (float*)d_out;      // out[0] = loss, out[1..] = logits [B, C]
    float* ws  = (float*)d_ws;       // ws[0] = contrastive sum, ws[1] = CE sum, ws[2..] = sq[B]
    float* sq  = ws + 2;

    init_kernel<<<1, 64, 0, stream>>>(ws);
    sq_kernel<<<BN / 8, 256, 0, stream>>>(target, sq);

    const int npairs = NTILES * (NTILES + 1) / 2;  // 2080 upper-triangular tile pairs
    gram_loss_kernel<<<npairs, 256, 0, stream>>>(target, sq, labels, ws + 0);

    xent_kernel<<<BN, 128, 0, stream>>>(pooled, W, bias, labels, out + 1, ws + 1);
    finalize_kernel<<<1, 32, 0, stream>>>(ws, out);
}